// GraphConv1d_35759897706671
// MI455X (gfx1250) — compile-verified
//
#include <hip/hip_runtime.h>
#include <hip/hip_bf16.h>

typedef __attribute__((ext_vector_type(2))) float v2f;
typedef __attribute__((ext_vector_type(8))) float v8f;

#define WMMA_F32(a, b, c) \
  __builtin_amdgcn_wmma_f32_16x16x4_f32(false, (a), false, (b), (short)0, (c), false, false)

static constexpr int BN = 8;     // batches
static constexpr int NN = 2048;  // nodes
static constexpr int FF = 128;   // features

// ---------------------------------------------------------------------------
// Kernel 1: row L2-normalize.  One wave per 128-float row.
// ---------------------------------------------------------------------------
__global__ __launch_bounds__(256) void normalize_rows(const float* __restrict__ h,
                                                      float* __restrict__ xn) {
  int row  = blockIdx.x * 8 + (threadIdx.x >> 5);
  int lane = threadIdx.x & 31;
  const float4* src = (const float4*)(h + (size_t)row * FF);
  float4 v = src[lane];
  float ss = v.x * v.x + v.y * v.y + v.z * v.z + v.w * v.w;
  ss += __shfl_xor(ss, 16, 32);
  ss += __shfl_xor(ss, 8, 32);
  ss += __shfl_xor(ss, 4, 32);
  ss += __shfl_xor(ss, 2, 32);
  ss += __shfl_xor(ss, 1, 32);
  float s = 1.0f / fmaxf(sqrtf(ss), 1e-8f);
  float4 o;
  o.x = v.x * s; o.y = v.y * s; o.z = v.z * s; o.w = v.w * s;
  ((float4*)(xn + (size_t)row * FF))[lane] = o;
}

// ---------------------------------------------------------------------------
// Kernel 2: thresholded-adjacency row sums -> invs = 1/sqrt(d).
// Grid: 128 blocks (16/batch) x 256 threads (8 waves). Wave owns 16 rows.
// ---------------------------------------------------------------------------
__global__ __launch_bounds__(256) void rowsum_kernel(const float* __restrict__ xn,
                                                     float* __restrict__ invs, float thr) {
  __shared__ float xs[16 * 132];  // m-tile of xn, padded pitch (bank-conflict free)
  int w = threadIdx.x >> 5, lane = threadIdx.x & 31;
  int l15 = lane & 15, hl = lane >> 4;
  int b = blockIdx.x >> 4;
  int row0 = (blockIdx.x & 15) * 128 + w * 16;
  const float* xb = xn + (size_t)b * NN * FF;

  // Resident A fragments: rows [row0..row0+15] x K=128 (32 k-steps of 4).
  v2f a[32];
  const float* arow = xb + (size_t)(row0 + l15) * FF + 2 * hl;
#pragma unroll
  for (int j = 0; j < 32; ++j) a[j] = *(const v2f*)(arow + 4 * j);

  float dacc[8] = {0, 0, 0, 0, 0, 0, 0, 0};
  for (int mt = 0; mt < NN / 16; ++mt) {
    __syncthreads();
    const float4* src = (const float4*)(xb + (size_t)mt * 16 * FF);
    for (int i = threadIdx.x; i < 512; i += 256)
      *(float4*)&xs[(i >> 5) * 132 + (i & 31) * 4] = src[i];
    __syncthreads();

    v8f c = {};
#pragma unroll
    for (int j = 0; j < 32; ++j) {
      int kb = 4 * j + 2 * hl;
      v2f bf = *(const v2f*)&xs[l15 * 132 + kb];  // B[k][n] = xn[m0+n][k]
      c = WMMA_F32(a[j], bf, c);
    }
#pragma unroll
    for (int v = 0; v < 8; ++v) {
      float val = c[v];
      dacc[v] += (val > thr) ? val : 0.0f;
    }
  }
  // Sum over the 16 columns held by the 16 lanes of each half-wave.
#pragma unroll
  for (int v = 0; v < 8; ++v) {
    float s = dacc[v];
    s += __shfl_xor(s, 1, 32);
    s += __shfl_xor(s, 2, 32);
    s += __shfl_xor(s, 4, 32);
    s += __shfl_xor(s, 8, 32);
    dacc[v] = s;
  }
  if (l15 == 0) {
#pragma unroll
    for (int v = 0; v < 8; ++v) {
      int r = row0 + v + 8 * hl;
      invs[(size_t)b * NN + r] = 1.0f / sqrtf(dacc[v]);
    }
  }
}

// ---------------------------------------------------------------------------
// Kernel 3: h_out = diag(invs) * thr(xn xn^T) * diag(invs) * h_in
// Adjacency tiles recomputed on the fly (no N*N materialization).
// ---------------------------------------------------------------------------
__global__ __launch_bounds__(256) void propagate_kernel(const float* __restrict__ xn,
                                                        const float* __restrict__ hin,
                                                        const float* __restrict__ invs,
                                                        float* __restrict__ hout, float thr) {
  __shared__ float xs[16 * 132];       // xn m-tile
  __shared__ float hs[16 * 132];       // h  m-tile
  __shared__ float adj[8][16 * 20];    // per-wave adjacency slab (pitch 20: no conflicts)
  __shared__ float ivs[NN];            // per-batch invs
  int w = threadIdx.x >> 5, lane = threadIdx.x & 31;
  int l15 = lane & 15, hl = lane >> 4;
  int b = blockIdx.x >> 4;
  int row0 = (blockIdx.x & 15) * 128 + w * 16;
  const float* xb = xn + (size_t)b * NN * FF;
  const float* hb = hin + (size_t)b * NN * FF;

  for (int i = threadIdx.x; i < NN; i += 256) ivs[i] = invs[(size_t)b * NN + i];

  v2f a[32];
  const float* arow = xb + (size_t)(row0 + l15) * FF + 2 * hl;
#pragma unroll
  for (int j = 0; j < 32; ++j) a[j] = *(const v2f*)(arow + 4 * j);

  v8f hacc[8] = {};
  for (int mt = 0; mt < NN / 16; ++mt) {
    __syncthreads();
    const float4* sx = (const float4*)(xb + (size_t)mt * 16 * FF);
    const float4* sh = (const float4*)(hb + (size_t)mt * 16 * FF);
    for (int i = threadIdx.x; i < 512; i += 256) {
      int d = (i >> 5) * 132 + (i & 31) * 4;
      *(float4*)&xs[d] = sx[i];
      *(float4*)&hs[d] = sh[i];
    }
    __syncthreads();

    // 16x16 adjacency tile, fp32 WMMA over K=128.
    v8f c = {};
#pragma unroll
    for (int j = 0; j < 32; ++j) {
      int kb = 4 * j + 2 * hl;
      v2f bf = *(const v2f*)&xs[l15 * 132 + kb];
      c = WMMA_F32(a[j], bf, c);
    }
    // threshold + column scale, park in LDS to re-shape D-layout -> A-layout.
    float cinv = ivs[mt * 16 + l15];
#pragma unroll
    for (int v = 0; v < 8; ++v) {
      float val = c[v];
      adj[w][(v + 8 * hl) * 20 + l15] = (val > thr) ? val * cinv : 0.0f;
    }
    __syncthreads();

    // hacc[ft] += adjTile(16x16) x h_tile(16x128)
#pragma unroll
    for (int ft = 0; ft < 8; ++ft) {
#pragma unroll
      for (int j = 0; j < 4; ++j) {
        int kb = 4 * j + 2 * hl;
        v2f a2 = *(const v2f*)&adj[w][l15 * 20 + kb];
        v2f b2;
        b2.x = hs[kb * 132 + ft * 16 + l15];
        b2.y = hs[(kb + 1) * 132 + ft * 16 + l15];
        hacc[ft] = WMMA_F32(a2, b2, hacc[ft]);
      }
    }
  }
  // epilogue: row scale + store
  float* ob = hout + (size_t)b * NN * FF;
#pragma unroll
  for (int ft = 0; ft < 8; ++ft) {
#pragma unroll
    for (int v = 0; v < 8; ++v) {
      int r = row0 + v + 8 * hl;
      ob[(size_t)r * FF + ft * 16 + l15] = hacc[ft][v] * ivs[r];
    }
  }
}

// ---------------------------------------------------------------------------
// Kernel 4: out = (h3 + res) @ W   (W staged whole in 64KB LDS)
// ---------------------------------------------------------------------------
__global__ __launch_bounds__(256) void final_gemm(const float* __restrict__ h3,
                                                  const float* __restrict__ res,
                                                  const float* __restrict__ W,
                                                  float* __restrict__ out) {
  __shared__ float wsm[128 * 128];
  int w = threadIdx.x >> 5, lane = threadIdx.x & 31;
  int l15 = lane & 15, hl = lane >> 4;
  int b = blockIdx.x >> 4;
  int row0 = (blockIdx.x & 15) * 128 + w * 16;

  for (int i = threadIdx.x * 4; i < 128 * 128; i += 256 * 4)
    *(float4*)&wsm[i] = *(const float4*)&W[i];

  const float* hrow = h3 + ((size_t)b * NN + row0 + l15) * FF + 2 * hl;
  const float* xrow = res + ((size_t)b * NN + row0 + l15) * FF + 2 * hl;
  v2f a[32];
#pragma unroll
  for (int j = 0; j < 32; ++j) {
    v2f ha = *(const v2f*)(hrow + 4 * j);
    v2f xa = *(const v2f*)(xrow + 4 * j);
    a[j] = ha + xa;
  }
  v8f acc[8] = {};
  __syncthreads();
#pragma unroll
  for (int ft = 0; ft < 8; ++ft) {
#pragma unroll
    for (int j = 0; j < 32; ++j) {
      int kb = 4 * j + 2 * hl;
      v2f bf;
      bf.x = wsm[kb * 128 + ft * 16 + l15];
      bf.y = wsm[(kb + 1) * 128 + ft * 16 + l15];
      acc[ft] = WMMA_F32(a[j], bf, acc[ft]);
    }
  }
#pragma unroll
  for (int ft = 0; ft < 8; ++ft) {
#pragma unroll
    for (int v = 0; v < 8; ++v) {
      int r = row0 + v + 8 * hl;
      out[((size_t)b * NN + r) * FF + ft * 16 + l15] = acc[ft][v];
    }
  }
}

// ---------------------------------------------------------------------------
extern "C" void kernel_launch(void* const* d_in, const int* in_sizes, int n_in,
                              void* d_out, int out_size, void* d_ws, size_t ws_size,
                              hipStream_t stream) {
  (void)in_sizes; (void)n_in; (void)out_size; (void)ws_size;
  const float* x = (const float*)d_in[0];
  const float* W = (const float*)d_in[1];
  float* out = (float*)d_out;

  char* ws = (char*)d_ws;
  const size_t tensorBytes = (size_t)BN * NN * FF * sizeof(float);  // 8 MB
  float* xn   = (float*)(ws);
  float* ha   = (float*)(ws + tensorBytes);
  float* hb   = (float*)(ws + 2 * tensorBytes);
  float* invs = (float*)(ws + 3 * tensorBytes);  // B*N floats

  const float thr[3] = {0.05f, 0.10f, 0.15f};
  float* houts[3] = {ha, hb, ha};
  const float* hin = x;
  for (int s = 0; s < 3; ++s) {
    normalize_rows<<<(BN * NN) / 8, 256, 0, stream>>>(hin, xn);
    rowsum_kernel<<<BN * (NN / 128), 256, 0, stream>>>(xn, invs, thr[s]);
    propagate_kernel<<<BN * (NN / 128), 256, 0, stream>>>(xn, hin, invs, houts[s], thr[s]);
    hin = houts[s];
  }
  final_gemm<<<BN * (NN / 128), 256, 0, stream>>>(hin, x, W, out);
}